// TreeGCN_10617159156362
// MI455X (gfx1250) — compile-verified
//
#include <hip/hip_runtime.h>

// ---------------------------------------------------------------------------
// TreeGCN layer for MI455X (gfx1250), fp32 via V_WMMA_F32_16X16X4_F32.
// Algebraic fold: (xb@Ws1)@Ws2 == xb@(Ws1@Ws2)  -> precompute Wss (64x64).
// Block = (node, 32-batch tile): branch[n] fragments loaded ONCE per block,
// shared across two 16-row M-tiles (two accumulators per B fragment).
// ---------------------------------------------------------------------------

typedef __attribute__((ext_vector_type(2))) float v2f;
typedef __attribute__((ext_vector_type(8))) float v8f;

#define B_     64
#define NODES_ 512
#define DEG_   8
#define INF_   64
#define OUTF_  64
#define SUP_   640

__device__ __forceinline__ v8f wmma4(v2f a, v2f b, v8f c) {
  // D = A(16x4 f32) x B(4x16 f32) + C(16x16 f32)
  return __builtin_amdgcn_wmma_f32_16x16x4_f32(
      /*neg_a=*/false, a, /*neg_b=*/false, b,
      /*c_mod=*/(short)0, c, /*reuse_a=*/false, /*reuse_b=*/false);
}

__device__ __forceinline__ float lrelu(float v) { return v > 0.0f ? v : 0.2f * v; }

// --------------------------- prologue kernels ------------------------------

// Wss = Ws1 (64x640) @ Ws2 (640x64)
__global__ void k_wss(const float* __restrict__ Ws1,
                      const float* __restrict__ Ws2,
                      float* __restrict__ Wss) {
  int idx = blockIdx.x * blockDim.x + threadIdx.x;  // 4096
  int i = idx >> 6, j = idx & 63;
  float acc = 0.0f;
  for (int k = 0; k < SUP_; ++k)
    acc = fmaf(Ws1[i * SUP_ + k], Ws2[k * 64 + j], acc);
  Wss[idx] = acc;
}

// tsum[b][j][f] = x0[b]@W0 + x1[b][j/8]@W1 + x2[b][j]@W2   (B x 64 x 64)
__global__ void k_tsum(const float* __restrict__ x0, const float* __restrict__ W0,
                       const float* __restrict__ x1, const float* __restrict__ W1,
                       const float* __restrict__ x2, const float* __restrict__ W2,
                       float* __restrict__ tsum) {
  int idx = blockIdx.x * blockDim.x + threadIdx.x;  // B*64*64 = 262144
  int f = idx & 63;
  int j = (idx >> 6) & 63;
  int b = idx >> 12;
  float acc = 0.0f;
  const float* xr0 = x0 + b * 96;
  for (int k = 0; k < 96; ++k) acc = fmaf(xr0[k], W0[k * 64 + f], acc);
  const float* xr1 = x1 + (b * 8 + (j >> 3)) * 64;
  for (int k = 0; k < 64; ++k) acc = fmaf(xr1[k], W1[k * 64 + f], acc);
  const float* xr2 = x2 + (b * 64 + j) * 64;
  for (int k = 0; k < 64; ++k) acc = fmaf(xr2[k], W2[k * 64 + f], acc);
  tsum[idx] = acc;
}

// ------------------------------ main kernel --------------------------------
// grid = NODES_ * 2 (batch tiles of 32); block = 128 threads = 4 wave32.
__launch_bounds__(128)
__global__ void k_main(const float* __restrict__ x3, const float* __restrict__ W3,
                       const float* __restrict__ branch, const float* __restrict__ bias,
                       const float* __restrict__ Wss, const float* __restrict__ tsum,
                       float* __restrict__ out) {
  const int n    = blockIdx.x >> 1;
  const int b0   = (blockIdx.x & 1) * 32;
  const int tid  = threadIdx.x;
  const int lane = tid & 31;
  const int wv   = tid >> 5;
  const int hl   = lane >> 4;   // half-wave: 0|1
  const int mr   = lane & 15;   // row (A) / col (B,C,D) within tile

  __shared__ float sA[32][68];      // x3 tile: 32 batches x 64 feats   (8.7 KB)
  __shared__ float sP[32][260];     // lrelu(A @ branch[n]) half: 32x256 (33 KB)
  __shared__ float sY[32][68];      // A @ W3 + tsum tile               (8.7 KB)
  __shared__ float sBias[DEG_ * OUTF_];                               // (2 KB)

  // cooperative loads
  for (int i = tid; i < 32 * 64; i += 128) {
    int r = i >> 6, c = i & 63;
    sA[r][c] = x3[((size_t)(b0 + r) * NODES_ + n) * INF_ + c];
  }
  for (int i = tid; i < DEG_ * OUTF_; i += 128) sBias[i] = bias[i];
  __syncthreads();

  const float* bp = branch + (size_t)n * (INF_ * INF_ * DEG_);  // 64 x 512

  // ---- Stage Y: Y = A @ W3 + tsum;  8 jobs (4 nt x 2 mt), 2 per wave ------
  for (int j = 0; j < 2; ++j) {
    const int job = wv * 2 + j;      // 0..7
    const int nt  = job >> 1;        // 0..3
    const int mt  = job & 1;         // 0..1
    const int col = nt * 16 + mr;
    v8f acc = {};
    for (int ks = 0; ks < 16; ++ks) {
      const int k0 = ks * 4 + hl * 2;
      v2f a, b;
      a.x = sA[mt * 16 + mr][k0];
      a.y = sA[mt * 16 + mr][k0 + 1];
      b.x = W3[k0 * 64 + col];          // small, cache-hot
      b.y = W3[(k0 + 1) * 64 + col];
      acc = wmma4(a, b, acc);
    }
#pragma unroll
    for (int v = 0; v < 8; ++v) {
      const int m = mt * 16 + v + hl * 8;
      sY[m][col] = acc[v] + tsum[((size_t)(b0 + m) * 64 + (n >> 3)) * 64 + col];
    }
  }
  __syncthreads();

  // ---- Two phases over N halves (256 cols each == 4 degrees) --------------
  for (int p = 0; p < 2; ++p) {
    const int c0 = p * 256;

    // Stage 1: P = lrelu(A @ branch[n][:, c0:c0+256])
    // wave wv owns N-tiles (p*16 + wv*4 .. +3); ONE B load feeds TWO M-tiles.
    for (int t = 0; t < 4; ++t) {
      const int ntl = wv * 4 + t;        // 0..15 within phase
      const int col = c0 + ntl * 16 + mr;
      const int lc  = ntl * 16 + mr;     // local column in sP
      v8f acc0 = {};
      v8f acc1 = {};
      for (int ks = 0; ks < 16; ++ks) {
        const int k0 = ks * 4 + hl * 2;
        v2f a0, a1, b;
        b.x  = bp[(size_t)k0 * 512 + col];
        b.y  = bp[(size_t)(k0 + 1) * 512 + col];
        a0.x = sA[mr][k0];
        a0.y = sA[mr][k0 + 1];
        a1.x = sA[16 + mr][k0];
        a1.y = sA[16 + mr][k0 + 1];
        acc0 = wmma4(a0, b, acc0);
        acc1 = wmma4(a1, b, acc1);
      }
#pragma unroll
      for (int v = 0; v < 8; ++v) {
        sP[v + hl * 8][lc]      = lrelu(acc0[v]);
        sP[16 + v + hl * 8][lc] = lrelu(acc1[v]);
      }
    }
    __syncthreads();

    // Stage 2: per (d, nt, mt): R = P_d @ Wss; out = lrelu(R + Y + bias)
    for (int j = 0; j < 8; ++j) {
      const int job = wv * 8 + j;        // 0..31
      const int dl  = job >> 3;          // 0..3 (local degree)
      const int nt  = (job >> 1) & 3;    // 0..3
      const int mt  = job & 1;           // 0..1
      const int d   = p * 4 + dl;
      const int col = nt * 16 + mr;      // output feature 0..63
      v8f acc = {};
      for (int ks = 0; ks < 16; ++ks) {
        const int k0 = ks * 4 + hl * 2;
        v2f a, b;
        a.x = sP[mt * 16 + mr][dl * 64 + k0];
        a.y = sP[mt * 16 + mr][dl * 64 + k0 + 1];
        b.x = Wss[k0 * 64 + col];        // 16 KB, cache-hot
        b.y = Wss[(k0 + 1) * 64 + col];
        acc = wmma4(a, b, acc);
      }
      const float bv = sBias[d * OUTF_ + col];
#pragma unroll
      for (int v = 0; v < 8; ++v) {
        const int m  = mt * 16 + v + hl * 8;
        const float val = lrelu(acc[v] + sY[m][col] + bv);
        float* dst = out + ((size_t)(b0 + m) * (NODES_ * DEG_) + (size_t)n * DEG_ + d) * OUTF_ + col;
        __builtin_nontemporal_store(val, dst);   // streamed once; keep L2 for branch
      }
    }
    __syncthreads();   // sP reused next phase
  }
}

// ------------------------------- launcher ----------------------------------

extern "C" void kernel_launch(void* const* d_in, const int* in_sizes, int n_in,
                              void* d_out, int out_size, void* d_ws, size_t ws_size,
                              hipStream_t stream) {
  (void)in_sizes; (void)n_in; (void)out_size; (void)ws_size;
  const float* x0     = (const float*)d_in[0];
  const float* W0     = (const float*)d_in[1];
  const float* x1     = (const float*)d_in[2];
  const float* W1     = (const float*)d_in[3];
  const float* x2     = (const float*)d_in[4];
  const float* W2     = (const float*)d_in[5];
  const float* x3     = (const float*)d_in[6];
  const float* W3     = (const float*)d_in[7];
  const float* branch = (const float*)d_in[8];
  const float* Ws1    = (const float*)d_in[9];
  const float* Ws2    = (const float*)d_in[10];
  const float* bias   = (const float*)d_in[11];
  float* out = (float*)d_out;

  float* Wss  = (float*)d_ws;            // 64*64 f32
  float* tsum = Wss + 64 * 64;           // B*64*64 f32 (~1 MB)

  k_wss<<<16, 256, 0, stream>>>(Ws1, Ws2, Wss);
  k_tsum<<<1024, 256, 0, stream>>>(x0, W0, x1, W1, x2, W2, tsum);
  k_main<<<NODES_ * 2, 128, 0, stream>>>(x3, W3, branch, bias, Wss, tsum, out);
}